// TactileGAT_43207370997900
// MI455X (gfx1250) — compile-verified
//
#include <hip/hip_runtime.h>
#include <stdint.h>

#define N_NODES 32000
#define N_EDGES 256000
#define N_GRAPH 64
#define DIN     64
#define TDIM    32
#define INTD    96      // DIN + TDIM
#define DMODEL  128
#define NHEADS  4
#define CHEAD   32
#define EDIM    16
#define NLAYER  4
#define FFDIM   512

#define LDSTRIDE 36     // padded LDS row stride (floats): 144B = 16B aligned, conflict-free

typedef __attribute__((ext_vector_type(2))) float v2f;
typedef __attribute__((ext_vector_type(8))) float v8f;

// ---------- helpers ----------
__device__ __forceinline__ float wred(float x) {
  #pragma unroll
  for (int off = 16; off; off >>= 1) x += __shfl_xor(x, off, 32);
  return x;
}
// order-preserving float<->uint encoding for atomicMax on floats
__device__ __forceinline__ unsigned fenc(float f) {
  unsigned u = __float_as_uint(f);
  return (u & 0x80000000u) ? ~u : (u | 0x80000000u);
}
__device__ __forceinline__ float fdec(unsigned u) {
  return (u & 0x80000000u) ? __uint_as_float(u & 0x7fffffffu) : __uint_as_float(~u);
}

// ---------- fp32 WMMA GEMM: C[M,Nn] = act(A[M,K] @ B[K,Nn] + bias) ----------
// block (256 thr = 8 waves) computes a 64x128 tile. A K-chunks (64x32) are staged
// into LDS via global_load_async_to_lds_b128 (ASYNCcnt); wave w owns cols [w*16,w*16+16)
// and all 4 row tiles, so each B fragment is reused across 4 V_WMMA_F32_16X16X4_F32.
// LDS offsets are derived from &As[...] (low 32 bits of the flat address = LDS byte
// offset) so the shared array escapes and its reads stay live (real ds_load_b64s).
__global__ void __launch_bounds__(256)
gemm_wmma_f32_kernel(const float* __restrict__ A, const float* __restrict__ B,
                     const float* __restrict__ bias, float* __restrict__ C,
                     int M, int K, int Nn, int relu) {
  __shared__ float As[64 * LDSTRIDE];

  const int tid  = threadIdx.x;
  const int lane = tid & 31;
  const int w    = tid >> 5;            // wave 0..7 -> 16-col slice
  const int half = lane >> 4;           // 0: lanes 0-15, 1: lanes 16-31
  const int idx  = lane & 15;

  const int rowBase = blockIdx.x * 64;
  const int col     = blockIdx.y * 128 + w * 16 + idx;

  const float* Ablk = A + (size_t)rowBase * K;

  v8f acc0 = {}, acc1 = {}, acc2 = {}, acc3 = {};

  for (int kc = 0; kc < K; kc += 32) {
    __syncthreads();                    // protect LDS from previous chunk's readers
    // stage 64x32 A chunk: 512 float4s, 2 per thread, coalesced 16B/lane
    #pragma unroll
    for (int u = 0; u < 2; ++u) {
      int i   = tid + u * 256;
      int row = i >> 3, k4 = (i & 7) << 2;
      const float* gp = Ablk + (size_t)row * K + kc + k4;
      // low 32 bits of the flat LDS address == wave-relative LDS byte offset;
      // ptrtoint also makes As escape so the asm's "memory" clobber aliases it.
      unsigned lo = (unsigned)(uintptr_t)&As[row * LDSTRIDE + k4];
      asm volatile("global_load_async_to_lds_b128 %0, %1, off"
                   :: "v"(lo), "v"(gp) : "memory");
    }
    asm volatile("s_wait_asynccnt 0" ::: "memory");
    __syncthreads();

    #pragma unroll
    for (int kk = 0; kk < 32; kk += 4) {
      const int k0 = kk + (half << 1);  // A/B frag: VGPR0={K0|K2}, VGPR1={K1|K3}
      v2f b;
      b[0] = B[(size_t)k0 * Nn + col];
      b[1] = B[(size_t)(k0 + 1) * Nn + col];
      v2f a0, a1, a2, a3;
      { float2 t0 = *(const float2*)&As[(0 * 16 + idx) * LDSTRIDE + k0]; a0[0] = t0.x; a0[1] = t0.y; }
      { float2 t1 = *(const float2*)&As[(1 * 16 + idx) * LDSTRIDE + k0]; a1[0] = t1.x; a1[1] = t1.y; }
      { float2 t2 = *(const float2*)&As[(2 * 16 + idx) * LDSTRIDE + k0]; a2[0] = t2.x; a2[1] = t2.y; }
      { float2 t3 = *(const float2*)&As[(3 * 16 + idx) * LDSTRIDE + k0]; a3[0] = t3.x; a3[1] = t3.y; }
      acc0 = __builtin_amdgcn_wmma_f32_16x16x4_f32(false, a0, false, b, (short)0, acc0, false, false);
      acc1 = __builtin_amdgcn_wmma_f32_16x16x4_f32(false, a1, false, b, (short)0, acc1, false, false);
      acc2 = __builtin_amdgcn_wmma_f32_16x16x4_f32(false, a2, false, b, (short)0, acc2, false, false);
      acc3 = __builtin_amdgcn_wmma_f32_16x16x4_f32(false, a3, false, b, (short)0, acc3, false, false);
    }
  }

  const float bv = bias ? bias[col] : 0.0f;
  v8f* accs[4] = {&acc0, &acc1, &acc2, &acc3};
  #pragma unroll
  for (int rt = 0; rt < 4; ++rt) {
    #pragma unroll
    for (int r = 0; r < 8; ++r) {       // C/D layout: VGPR r -> M = r + 8*half
      float o = (*accs[rt])[r] + bv;
      if (relu) o = fmaxf(o, 0.0f);
      C[(size_t)(rowBase + rt * 16 + half * 8 + r) * Nn + col] = o;
    }
  }
}

// ---------- input concat: [x | sin(t*fac) | cos(t*fac)] ----------
__global__ void build_hin_kernel(const float* __restrict__ x, const int* __restrict__ t,
                                 float* __restrict__ hin) {
  int i = blockIdx.x * blockDim.x + threadIdx.x;
  if (i >= N_NODES * INTD) return;
  int n = i / INTD, c = i % INTD;
  float v;
  if (c < DIN) {
    v = x[n * DIN + c];
  } else {
    int j = (c - DIN) & 15;
    float fac = expf(-9.210340371976184f * (float)j / 15.00000001f);
    float ang = (float)t[n] * fac;
    v = (c < DIN + 16) ? sinf(ang) : cosf(ang);
  }
  hin[i] = v;
}

// ---------- per-layer attention scratch init ----------
__global__ void attn_init_kernel(float* __restrict__ out, unsigned* __restrict__ amax,
                                 float* __restrict__ den) {
  int i = blockIdx.x * blockDim.x + threadIdx.x;
  if (i < N_NODES * DMODEL) out[i] = 0.0f;
  if (i < N_NODES * NHEADS) { amax[i] = fenc(-1e30f); den[i] = 0.0f; }
}

// ---------- pass 1: alpha + segment max (wave per edge; edge proj recomputed on the fly) ----------
__global__ void attn_alpha_kernel(const int* __restrict__ src, const int* __restrict__ dst,
                                  const float* __restrict__ q, const float* __restrict__ k,
                                  const float* __restrict__ ea, const float* __restrict__ We,
                                  const float* __restrict__ be,
                                  float* __restrict__ alpha, unsigned* __restrict__ amax) {
  int lane = threadIdx.x & 31;
  int e = blockIdx.x * (blockDim.x >> 5) + (threadIdx.x >> 5);
  if (e >= N_EDGES) return;
  int s = src[e], d = dst[e];
  float ar[EDIM];
  #pragma unroll
  for (int j = 0; j < EDIM; ++j) ar[j] = ea[(size_t)e * EDIM + j];
  #pragma unroll
  for (int hh = 0; hh < NHEADS; ++hh) {
    int c = hh * CHEAD + lane;
    float ep = be[c];
    #pragma unroll
    for (int j = 0; j < EDIM; ++j) ep += ar[j] * We[j * DMODEL + c];
    float val = q[(size_t)d * DMODEL + c] * (k[(size_t)s * DMODEL + c] + ep);
    float sum = wred(val) * 0.17677669529663687f;   // 1/sqrt(C)
    if (lane == 0) {
      alpha[e * NHEADS + hh] = sum;
      atomicMax(&amax[d * NHEADS + hh], fenc(sum));
    }
  }
}

// ---------- pass 2: exp(alpha - max), segment sum ----------
__global__ void attn_ex_kernel(const int* __restrict__ dst, float* __restrict__ alpha,
                               const unsigned* __restrict__ amax, float* __restrict__ den) {
  int i = blockIdx.x * blockDim.x + threadIdx.x;
  if (i >= N_EDGES * NHEADS) return;
  int e = i >> 2, hh = i & 3;
  int d = dst[e];
  float ex = expf(alpha[i] - fdec(amax[d * NHEADS + hh]));
  alpha[i] = ex;
  atomicAdd(&den[d * NHEADS + hh], ex);
}

// ---------- pass 3: weighted value scatter (wave per edge) ----------
__global__ void attn_scatter_kernel(const int* __restrict__ src, const int* __restrict__ dst,
                                    const float* __restrict__ v, const float* __restrict__ ea,
                                    const float* __restrict__ We, const float* __restrict__ be,
                                    const float* __restrict__ exb, const float* __restrict__ den,
                                    float* __restrict__ out) {
  int lane = threadIdx.x & 31;
  int e = blockIdx.x * (blockDim.x >> 5) + (threadIdx.x >> 5);
  if (e >= N_EDGES) return;
  int s = src[e], d = dst[e];
  float ar[EDIM];
  #pragma unroll
  for (int j = 0; j < EDIM; ++j) ar[j] = ea[(size_t)e * EDIM + j];
  #pragma unroll
  for (int hh = 0; hh < NHEADS; ++hh) {
    int c = hh * CHEAD + lane;
    float w = exb[e * NHEADS + hh] / den[d * NHEADS + hh];
    float ep = be[c];
    #pragma unroll
    for (int j = 0; j < EDIM; ++j) ep += ar[j] * We[j * DMODEL + c];
    atomicAdd(&out[(size_t)d * DMODEL + c], (v[(size_t)s * DMODEL + c] + ep) * w);
  }
}

// ---------- beta gate + residual + LayerNorm1 (wave per node, 4 elems/lane) ----------
__global__ void beta_ln_kernel(float* __restrict__ h, const float* __restrict__ out,
                               const float* __restrict__ xr, const float* __restrict__ Wb,
                               const float* __restrict__ g, const float* __restrict__ b) {
  int lane = threadIdx.x & 31;
  int n = blockIdx.x * (blockDim.x >> 5) + (threadIdx.x >> 5);
  if (n >= N_NODES) return;
  size_t base = (size_t)n * DMODEL;
  float ov[4], xv[4], hv[4];
  float bd = 0.0f;
  #pragma unroll
  for (int j = 0; j < 4; ++j) {
    int dd = j * 32 + lane;
    ov[j] = out[base + dd]; xv[j] = xr[base + dd]; hv[j] = h[base + dd];
    bd += ov[j] * Wb[dd] + xv[j] * Wb[DMODEL + dd] + (ov[j] - xv[j]) * Wb[2 * DMODEL + dd];
  }
  float beta = 1.0f / (1.0f + expf(-wred(bd)));
  float y[4]; float s = 0.0f;
  #pragma unroll
  for (int j = 0; j < 4; ++j) { y[j] = hv[j] + beta * xv[j] + (1.0f - beta) * ov[j]; s += y[j]; }
  float mean = wred(s) * (1.0f / DMODEL);
  float vs = 0.0f;
  #pragma unroll
  for (int j = 0; j < 4; ++j) { float dlt = y[j] - mean; vs += dlt * dlt; }
  float inv = rsqrtf(wred(vs) * (1.0f / DMODEL) + 1e-5f);
  #pragma unroll
  for (int j = 0; j < 4; ++j) {
    int dd = j * 32 + lane;
    h[base + dd] = (y[j] - mean) * inv * g[dd] + b[dd];
  }
}

// ---------- residual + LayerNorm2 ----------
__global__ void add_ln_kernel(float* __restrict__ h, const float* __restrict__ add,
                              const float* __restrict__ g, const float* __restrict__ b) {
  int lane = threadIdx.x & 31;
  int n = blockIdx.x * (blockDim.x >> 5) + (threadIdx.x >> 5);
  if (n >= N_NODES) return;
  size_t base = (size_t)n * DMODEL;
  float y[4]; float s = 0.0f;
  #pragma unroll
  for (int j = 0; j < 4; ++j) { int dd = j * 32 + lane; y[j] = h[base + dd] + add[base + dd]; s += y[j]; }
  float mean = wred(s) * (1.0f / DMODEL);
  float vs = 0.0f;
  #pragma unroll
  for (int j = 0; j < 4; ++j) { float dlt = y[j] - mean; vs += dlt * dlt; }
  float inv = rsqrtf(wred(vs) * (1.0f / DMODEL) + 1e-5f);
  #pragma unroll
  for (int j = 0; j < 4; ++j) { int dd = j * 32 + lane; h[base + dd] = (y[j] - mean) * inv * g[dd] + b[dd]; }
}

// ---------- readout ----------
__global__ void readout_init_kernel(int* __restrict__ tmax, float* __restrict__ cnt,
                                    float* __restrict__ gsum, unsigned* __restrict__ gmax) {
  int i = blockIdx.x * blockDim.x + threadIdx.x;
  if (i < N_GRAPH) { tmax[i] = -1; cnt[i] = 0.0f; }
  if (i < N_GRAPH * DMODEL) { gsum[i] = 0.0f; gmax[i] = fenc(-1e30f); }
}

__global__ void tmax_kernel(const int* __restrict__ batch, const int* __restrict__ t,
                            int* __restrict__ tmax) {
  int n = blockIdx.x * blockDim.x + threadIdx.x;
  if (n >= N_NODES) return;
  atomicMax(&tmax[batch[n]], t[n]);
}

__global__ void readout_accum_kernel(const int* __restrict__ batch, const int* __restrict__ t,
                                     const int* __restrict__ tmax, const float* __restrict__ h,
                                     float* __restrict__ cnt, float* __restrict__ gsum,
                                     unsigned* __restrict__ gmax) {
  int lane = threadIdx.x & 31;
  int n = blockIdx.x * (blockDim.x >> 5) + (threadIdx.x >> 5);
  if (n >= N_NODES) return;
  int g = batch[n];
  if (t[n] != tmax[g]) return;
  if (lane == 0) atomicAdd(&cnt[g], 1.0f);
  #pragma unroll
  for (int j = 0; j < 4; ++j) {
    int dd = j * 32 + lane;
    float hv = h[(size_t)n * DMODEL + dd];
    atomicAdd(&gsum[g * DMODEL + dd], hv);
    atomicMax(&gmax[g * DMODEL + dd], fenc(hv));
  }
}

__global__ void head_kernel(const float* __restrict__ gsum, const unsigned* __restrict__ gmax,
                            const float* __restrict__ cnt, const float* __restrict__ Wh1,
                            const float* __restrict__ bh1, const float* __restrict__ Wh2,
                            const float* __restrict__ bh2, float* __restrict__ out) {
  __shared__ float hid[DMODEL];
  int g = blockIdx.x, d = threadIdx.x;
  float c = fmaxf(cnt[g], 1.0f);
  float acc = bh1[d];
  for (int j = 0; j < DMODEL; ++j) acc += (gsum[g * DMODEL + j] / c) * Wh1[j * DMODEL + d];
  for (int j = 0; j < DMODEL; ++j) acc += fdec(gmax[g * DMODEL + j]) * Wh1[(DMODEL + j) * DMODEL + d];
  hid[d] = fmaxf(acc, 0.0f);
  __syncthreads();
  if (d < 2) {
    float s = bh2[d];
    for (int j = 0; j < DMODEL; ++j) s += hid[j] * Wh2[j * 2 + d];
    out[g * 2 + d] = s;
  }
}

// ---------- host ----------
extern "C" void kernel_launch(void* const* d_in, const int* in_sizes, int n_in,
                              void* d_out, int out_size, void* d_ws, size_t ws_size,
                              hipStream_t stream) {
  const float* x      = (const float*)d_in[0];
  const int*   t      = (const int*)d_in[1];
  const int*   eidx   = (const int*)d_in[2];
  const float* eattr  = (const float*)d_in[3];
  const int*   batch  = (const int*)d_in[4];
  const float* Win    = (const float*)d_in[5];
  const float* b_in   = (const float*)d_in[6];
  const float* Wq     = (const float*)d_in[7];
  const float* bq     = (const float*)d_in[8];
  const float* Wk     = (const float*)d_in[9];
  const float* bk     = (const float*)d_in[10];
  const float* Wv     = (const float*)d_in[11];
  const float* bv     = (const float*)d_in[12];
  const float* We     = (const float*)d_in[13];
  const float* be     = (const float*)d_in[14];
  const float* Wskip  = (const float*)d_in[15];
  const float* bskip  = (const float*)d_in[16];
  const float* Wbeta  = (const float*)d_in[17];
  const float* ln1g   = (const float*)d_in[18];
  const float* ln1b   = (const float*)d_in[19];
  const float* Wff1   = (const float*)d_in[20];
  const float* bff1   = (const float*)d_in[21];
  const float* Wff2   = (const float*)d_in[22];
  const float* bff2   = (const float*)d_in[23];
  const float* ln2g   = (const float*)d_in[24];
  const float* ln2b   = (const float*)d_in[25];
  const float* Wh1    = (const float*)d_in[26];
  const float* bh1    = (const float*)d_in[27];
  const float* Wh2    = (const float*)d_in[28];
  const float* bh2    = (const float*)d_in[29];

  const int* src = eidx;
  const int* dst = eidx + N_EDGES;

  // workspace carving (~170 MB)
  float* h      = (float*)d_ws;                      // N*128
  float* q      = h      + (size_t)N_NODES * DMODEL;
  float* kbuf   = q      + (size_t)N_NODES * DMODEL;
  float* vbuf   = kbuf   + (size_t)N_NODES * DMODEL;
  float* sk     = vbuf   + (size_t)N_NODES * DMODEL;
  float* tmp128 = sk     + (size_t)N_NODES * DMODEL; // attn out, then ff2
  float* albuf  = tmp128 + (size_t)N_NODES * DMODEL; // E*4  (alpha, then exp)
  float* den    = albuf  + (size_t)N_EDGES * NHEADS; // N*4
  unsigned* amax = (unsigned*)(den + (size_t)N_NODES * NHEADS);
  int*   tmaxb  = (int*)(amax + (size_t)N_NODES * NHEADS);
  float* cnt    = (float*)(tmaxb + N_GRAPH);
  float* gsum   = cnt + N_GRAPH;
  unsigned* gmax = (unsigned*)(gsum + N_GRAPH * DMODEL);
  float* big    = (float*)(gmax + N_GRAPH * DMODEL); // N*512 (hin, then ff1)

  auto gemm = [&](const float* A, const float* B, const float* bias, float* C,
                  int M, int K, int Nn, int relu) {
    dim3 grid(M / 64, Nn / 128);
    gemm_wmma_f32_kernel<<<grid, 256, 0, stream>>>(A, B, bias, C, M, K, Nn, relu);
  };

  // input projection
  build_hin_kernel<<<(N_NODES * INTD + 255) / 256, 256, 0, stream>>>(x, t, big);
  gemm(big, Win, b_in, h, N_NODES, INTD, DMODEL, 1);

  const int ewaves = (N_EDGES + 3) / 4;
  const int nwaves = (N_NODES + 3) / 4;

  for (int i = 0; i < NLAYER; ++i) {
    const float* Wq_i = Wq + (size_t)i * DMODEL * DMODEL;
    const float* Wk_i = Wk + (size_t)i * DMODEL * DMODEL;
    const float* Wv_i = Wv + (size_t)i * DMODEL * DMODEL;
    const float* Ws_i = Wskip + (size_t)i * DMODEL * DMODEL;
    const float* We_i = We + (size_t)i * EDIM * DMODEL;
    const float* W1_i = Wff1 + (size_t)i * DMODEL * FFDIM;
    const float* W2_i = Wff2 + (size_t)i * FFDIM * DMODEL;

    gemm(h, Wq_i, bq + i * DMODEL, q,    N_NODES, DMODEL, DMODEL, 0);
    gemm(h, Wk_i, bk + i * DMODEL, kbuf, N_NODES, DMODEL, DMODEL, 0);
    gemm(h, Wv_i, bv + i * DMODEL, vbuf, N_NODES, DMODEL, DMODEL, 0);
    gemm(h, Ws_i, bskip + i * DMODEL, sk, N_NODES, DMODEL, DMODEL, 0);

    attn_init_kernel<<<(N_NODES * DMODEL + 255) / 256, 256, 0, stream>>>(tmp128, amax, den);
    attn_alpha_kernel<<<ewaves, 128, 0, stream>>>(src, dst, q, kbuf, eattr, We_i,
                                                  be + i * DMODEL, albuf, amax);
    attn_ex_kernel<<<(N_EDGES * NHEADS + 255) / 256, 256, 0, stream>>>(dst, albuf, amax, den);
    attn_scatter_kernel<<<ewaves, 128, 0, stream>>>(src, dst, vbuf, eattr, We_i,
                                                    be + i * DMODEL, albuf, den, tmp128);
    beta_ln_kernel<<<nwaves, 128, 0, stream>>>(h, tmp128, sk, Wbeta + i * 3 * DMODEL,
                                               ln1g + i * DMODEL, ln1b + i * DMODEL);

    gemm(h, W1_i, bff1 + i * FFDIM, big, N_NODES, DMODEL, FFDIM, 1);
    gemm(big, W2_i, bff2 + i * DMODEL, tmp128, N_NODES, FFDIM, DMODEL, 0);
    add_ln_kernel<<<nwaves, 128, 0, stream>>>(h, tmp128, ln2g + i * DMODEL, ln2b + i * DMODEL);
  }

  // masked readout
  readout_init_kernel<<<(N_GRAPH * DMODEL + 255) / 256, 256, 0, stream>>>(tmaxb, cnt, gsum, gmax);
  tmax_kernel<<<(N_NODES + 255) / 256, 256, 0, stream>>>(batch, t, tmaxb);
  readout_accum_kernel<<<nwaves, 128, 0, stream>>>(batch, t, tmaxb, h, cnt, gsum, gmax);
  head_kernel<<<N_GRAPH, DMODEL, 0, stream>>>(gsum, gmax, cnt, Wh1, bh1, Wh2, bh2, (float*)d_out);
}